// LearnableEMA_57621281243424
// MI455X (gfx1250) — compile-verified
//
#include <hip/hip_runtime.h>
#include <hip/hip_bf16.h>

// ---------------------------------------------------------------------------
// Learnable EMA  y_t = a*y_{t-1} + (1-a)*x_t,  y_0 = x_0,  a = sigmoid(la[d])
// x: [B=32, T=4096, D=256] fp32   ->   y same shape.
// Chunked parallel scan, chunk = 16 timesteps:
//   K1: chunk-local terminal values e_c        (read x, write 8MB summaries)
//   K2: scan summaries -> carry-in per chunk   (in-place, tiny)
//   K3: WMMA tile apply: 16x16 prefix tile via U @ W with
//       U = shared lower-triangular ones (A operand),
//       W[k,d] = a_d^{-k} * z[k,d]       (B operand, features across lanes)
//       y[i,d] = a^i * (U@W)[i,d] + a^{i+1} * carry
// ---------------------------------------------------------------------------

typedef float v2f __attribute__((ext_vector_type(2)));
typedef float v8f __attribute__((ext_vector_type(8)));

#define B_      32
#define T_      4096
#define D_      256
#define CHUNK   16
#define NCHUNK  (T_ / CHUNK)   // 256

__device__ __forceinline__ float sigmoidf_(float x) {
  return 1.0f / (1.0f + __expf(-x));
}

// --------------------------------------------------------------------------
// K1: chunk summaries. 65536 threads; each thread owns 4 consecutive features
// (float4 loads -> 512B per wave access, fully coalesced) and 128 timesteps
// (8 chunks) of one batch. e[b, c, d] = sum_j a^{15-j} z_{16c+j}.
// --------------------------------------------------------------------------
__global__ void __launch_bounds__(256) ema_k1_summaries(
    const float* __restrict__ x, const float* __restrict__ la,
    float* __restrict__ e) {
  int tid  = blockIdx.x * 256 + threadIdx.x;     // 0 .. 65535
  int dq   = tid & 63;                           // 64 feature-quads
  int tseg = (tid >> 6) & 31;                    // 32 segments of 128 t
  int b    = tid >> 11;                          // batch
  int d    = dq * 4;

  float4 l4 = *(const float4*)(la + d);
  float a0 = sigmoidf_(l4.x), a1 = sigmoidf_(l4.y);
  float a2 = sigmoidf_(l4.z), a3 = sigmoidf_(l4.w);
  float o0 = 1.0f - a0, o1 = 1.0f - a1, o2 = 1.0f - a2, o3 = 1.0f - a3;

  int t = tseg * 128;
  for (int cl = 0; cl < 8; ++cl) {
    float s0 = 0.f, s1 = 0.f, s2 = 0.f, s3 = 0.f;
    #pragma unroll
    for (int j = 0; j < 16; ++j, ++t) {
      float4 xv = *(const float4*)(x + ((size_t)b * T_ + t) * D_ + d);
      // z = (t==0) ? x : (1-a)*x      (y_0 = x_0 boundary)
      float m0 = (t == 0) ? 1.0f : o0;
      float m1 = (t == 0) ? 1.0f : o1;
      float m2 = (t == 0) ? 1.0f : o2;
      float m3 = (t == 0) ? 1.0f : o3;
      s0 = fmaf(a0, s0, m0 * xv.x);
      s1 = fmaf(a1, s1, m1 * xv.y);
      s2 = fmaf(a2, s2, m2 * xv.z);
      s3 = fmaf(a3, s3, m3 * xv.w);
    }
    int c = tseg * 8 + cl;
    float4 sv = make_float4(s0, s1, s2, s3);
    *(float4*)(e + ((size_t)b * NCHUNK + c) * D_ + d) = sv;
  }
}

// --------------------------------------------------------------------------
// K2: in-place scan of summaries along chunks; leaves carry-IN per chunk:
//   tmp = e[c]; e[c] = h; h = a^16 * h + tmp
// 8192 threads, lanes span consecutive d -> coalesced.
// --------------------------------------------------------------------------
__global__ void __launch_bounds__(256) ema_k2_carry(
    const float* __restrict__ la, float* __restrict__ e) {
  int tid = blockIdx.x * 256 + threadIdx.x;      // 0 .. 8191
  int d = tid & (D_ - 1);
  int b = tid >> 8;
  float a  = sigmoidf_(la[d]);
  float p2 = a * a, p4 = p2 * p2, p8 = p4 * p4;
  float beta = p8 * p8;                          // a^16
  float h = 0.f;
  for (int c = 0; c < NCHUNK; ++c) {
    size_t idx = ((size_t)b * NCHUNK + c) * D_ + d;
    float tmp = e[idx];
    e[idx] = h;                                  // carry-in for chunk c
    h = fmaf(beta, h, tmp);
  }
}

// --------------------------------------------------------------------------
// K3: one wave = one (b, chunk, 16-feature tile). 131072 waves.
// acc[16x16] = U @ W via 4 chained v_wmma_f32_16x16x4_f32 (A = shared
// triangular ones; B = a^{-k}-scaled z tile, features across lanes).
// y[i,d] = a^i * (acc[i,d] + a*carry).
// --------------------------------------------------------------------------
__global__ void __launch_bounds__(256) ema_k3_apply(
    const float* __restrict__ x, const float* __restrict__ la,
    const float* __restrict__ carry, float* __restrict__ y) {
  int lane = threadIdx.x & 31;
  int wave = blockIdx.x * 8 + (threadIdx.x >> 5);
  int ft = wave & 15;                 // feature tile (16 features)
  int c  = (wave >> 4) & (NCHUNK - 1);
  int b  = wave >> 12;
  bool lower = lane < 16;
  int  m = lane & 15;                 // matrix row (A) / feature col (B,D)
  int  d = ft * 16 + m;

  float lav = la[d];
  float a  = sigmoidf_(lav);
  float om = 1.0f - a;
  float r  = 1.0f + __expf(-lav);     // exactly 1/sigmoid(la)
  float r2 = r * r, r3 = r2 * r, r4 = r2 * r2;

  const float* xb = x + ((size_t)b * T_ + (size_t)c * CHUNK) * D_ + d;

  int   kc0 = lower ? 0 : 2;          // K index held in B/A vgpr 0
  int   kc1 = lower ? 1 : 3;          // K index held in B/A vgpr 1
  float w0x = lower ? 1.0f : r2;      // a^{-kc} factors
  float w1x = lower ? r    : r3;

  v8f acc = {};
  float f = 1.0f;                     // a^{-4s}
  #pragma unroll
  for (int s = 0; s < 4; ++s) {
    int k0 = 4 * s + kc0;
    int k1 = 4 * s + kc1;
    float x0 = xb[(size_t)k0 * D_];
    float x1 = xb[(size_t)k1 * D_];
    // z: (1-a)*x, except the very first element of the whole sequence
    float z0 = ((c == 0) && (k0 == 0)) ? x0 : om * x0;
    float z1 = om * x1;
    v2f Bv; Bv.x = z0 * f * w0x; Bv.y = z1 * f * w1x;
    v2f Av; Av.x = (k0 <= m) ? 1.0f : 0.0f;   // U[m,k] = (k <= m)
            Av.y = (k1 <= m) ? 1.0f : 0.0f;
    acc = __builtin_amdgcn_wmma_f32_16x16x4_f32(
        /*neg_a=*/false, Av, /*neg_b=*/false, Bv,
        /*c_mod=*/(short)0, acc, /*reuse_a=*/false, /*reuse_b=*/false);
    f *= r4;
  }

  float hc = carry[((size_t)b * NCHUNK + c) * D_ + d];
  float hh = a * hc;                  // a^{M+1}h = a^M * (a*h)
  float p2 = a * a, p4 = p2 * p2, a8 = p4 * p4;
  float pm = lower ? 1.0f : a8;       // a^M, M = rI + (lower?0:8)

  float* yb = y + ((size_t)b * T_ + (size_t)c * CHUNK + (lower ? 0 : 8)) * D_ + d;
  #pragma unroll
  for (int rI = 0; rI < 8; ++rI) {
    yb[(size_t)rI * D_] = pm * (acc[rI] + hh);
    pm *= a;
  }
}

// --------------------------------------------------------------------------
extern "C" void kernel_launch(void* const* d_in, const int* in_sizes, int n_in,
                              void* d_out, int out_size, void* d_ws, size_t ws_size,
                              hipStream_t stream) {
  const float* x  = (const float*)d_in[0];
  const float* la = (const float*)d_in[1];
  float* y = (float*)d_out;
  float* e = (float*)d_ws;            // B*NCHUNK*D floats = 8 MB

  ema_k1_summaries<<<256, 256, 0, stream>>>(x, la, e);
  ema_k2_carry<<<32, 256, 0, stream>>>(la, e);
  ema_k3_apply<<<16384, 256, 0, stream>>>(x, la, e, y);
}